// SubExtractor_2972117369039
// MI455X (gfx1250) — compile-verified
//
#include <hip/hip_runtime.h>
#include <math.h>

// Problem constants (reference: N=131072, B=256, D=256, C=32, MAX_LEN=1024)
#define D_DIM 256
#define C_DIM 32
#define B_DIM 256
#define L_DIM 1024
#define SCALE_F 0.0625f   // 1/sqrt(256)

typedef __attribute__((ext_vector_type(2))) float v2f;
typedef __attribute__((ext_vector_type(8))) float v8f;
typedef __attribute__((ext_vector_type(4))) unsigned int v4u;
typedef __attribute__((ext_vector_type(8))) unsigned int v8u;

// D(16x16 f32) = A(16x4 f32) * B(4x16 f32) + C   -- CDNA5 V_WMMA_F32_16X16X4_F32
__device__ static inline v8f wmma4(v2f a, v2f b, v8f c) {
  return __builtin_amdgcn_wmma_f32_16x16x4_f32(
      /*neg_a=*/false, a, /*neg_b=*/false, b,
      /*c_mod=*/(short)0, c, /*reuse_a=*/false, /*reuse_b=*/false);
}

// ---- Tensor Data Mover: 2D tile (global -> LDS), D# per CDNA5 ISA ----
// All dims/strides in 4-byte elements. Rows/cols past tensor_dim* read as 0.
__device__ static inline void tdm_load_2d(unsigned int lds_addr, unsigned long long gaddr,
                                          unsigned int tensor_d0, unsigned int tensor_d1,
                                          unsigned int tile_d0, unsigned int tile_d1,
                                          unsigned int stride0) {
  v4u g0;
  g0[0] = 1u;                                        // count=1, user mode, no gather
  g0[1] = lds_addr;                                  // LDS byte address
  g0[2] = (unsigned int)gaddr;                       // global_addr[31:0]
  g0[3] = ((unsigned int)(gaddr >> 32) & 0x01FFFFFFu)
          | 0x80000000u;                             // global_addr[56:32] | type=2 (bits 127:126)
  v8u g1;
  g1[0] = 2u << 16;                                  // data_size=4B; workgroup_mask=0
  g1[1] = (tensor_d0 & 0xFFFFu) << 16;               // tensor_dim0[15:0] at bit 48
  g1[2] = (tensor_d0 >> 16) | ((tensor_d1 & 0xFFFFu) << 16);  // dim0[31:16] | dim1[15:0]
  g1[3] = (tensor_d1 >> 16) | (tile_d0 << 16);       // dim1[31:16] | tile_dim0
  g1[4] = tile_d1;                                   // tile_dim1; tile_dim2=0 (2D)
  g1[5] = stride0;                                   // tensor_dim0_stride[31:0]
  g1[6] = 0;                                         // stride0[47:32]=0; dim1_stride lo=0
  g1[7] = 0;
  asm volatile("tensor_load_to_lds %0, %1" :: "s"(g0), "s"(g1) : "memory");
}

// ---------------- segment bookkeeping ----------------
__global__ void k_zero_counts(int* counts) { counts[threadIdx.x] = 0; }

__global__ void k_hist(const int* __restrict__ batch, int* counts, int n) {
  int i = blockIdx.x * blockDim.x + threadIdx.x;
  if (i < n) atomicAdd(&counts[batch[i]], 1);
}

__global__ void k_scan(const int* __restrict__ counts, int* __restrict__ starts) {
  __shared__ int s[B_DIM];
  int t = threadIdx.x;
  s[t] = counts[t];
  __syncthreads();
  for (int off = 1; off < B_DIM; off <<= 1) {
    int add = (t >= off) ? s[t - off] : 0;
    __syncthreads();
    s[t] += add;
    __syncthreads();
  }
  starts[t] = s[t] - counts[t];   // exclusive prefix sum
}

// ---------------- tiny precomputes ----------------
// Qp[c,d] = sum_e Q[c,e]*Wq[d,e] + bq[d]
__global__ void k_qp(const float* __restrict__ Q, const float* __restrict__ Wq,
                     const float* __restrict__ bq, float* __restrict__ qp) {
  __shared__ float sq[D_DIM];
  int c = blockIdx.x, d = threadIdx.x;
  sq[d] = Q[c * D_DIM + d];
  __syncthreads();
  float acc = bq[d];
  for (int e = 0; e < D_DIM; ++e) acc += sq[e] * Wq[d * D_DIM + e];
  qp[c * D_DIM + d] = acc;
}

// QKs[c,e] = SCALE * sum_d Qp[c,d]*Wk[d,e];  qb[c] = SCALE * Qp[c].bk
__global__ void k_qk(const float* __restrict__ qp, const float* __restrict__ Wk,
                     const float* __restrict__ bk, float* __restrict__ qks,
                     float* __restrict__ qbv) {
  __shared__ float sq[D_DIM];
  int c = blockIdx.x, e = threadIdx.x;
  sq[e] = qp[c * D_DIM + e];
  __syncthreads();
  float acc = 0.f;
  for (int d = 0; d < D_DIM; ++d) acc += sq[d] * Wk[d * D_DIM + e];
  qks[c * D_DIM + e] = acc * SCALE_F;
  if (e == 0) {
    float s = 0.f;
    for (int d = 0; d < D_DIM; ++d) s += sq[d] * bk[d];
    qbv[c] = s * SCALE_F;
  }
}

// c2[c,d] = sum_e Wo[d,e]*(Qp[c,e]+bv[e]) + bo[d]
__global__ void k_c2(const float* __restrict__ qp, const float* __restrict__ Wo,
                     const float* __restrict__ bv, const float* __restrict__ bo,
                     float* __restrict__ c2) {
  __shared__ float sq[D_DIM];
  int c = blockIdx.x, d = threadIdx.x;
  sq[d] = qp[c * D_DIM + d] + bv[d];
  __syncthreads();
  float acc = bo[d];
  for (int e = 0; e < D_DIM; ++e) acc += Wo[d * D_DIM + e] * sq[e];
  c2[c * D_DIM + d] = acc;
}

// W2 = Wo @ Wv  (256x256x256), one wave per 16x16 tile, fp32 WMMA
__global__ void k_w2(const float* __restrict__ Wo, const float* __restrict__ Wv,
                     float* __restrict__ W2) {
  int L = threadIdx.x;
  int d0 = blockIdx.x * 16, e0 = blockIdx.y * 16;
  int m = L & 15, half = L >> 4, kb = half * 2;
  v8f acc = {};
  for (int kk = 0; kk < D_DIM; kk += 4) {
    v2f a, b;
    a.x = Wo[(d0 + m) * D_DIM + kk + kb];
    a.y = Wo[(d0 + m) * D_DIM + kk + kb + 1];
    b.x = Wv[(kk + kb) * D_DIM + e0 + m];
    b.y = Wv[(kk + kb + 1) * D_DIM + e0 + m];
    acc = wmma4(a, b, acc);
  }
  for (int r = 0; r < 8; ++r)
    W2[(d0 + r + half * 8) * D_DIM + e0 + m] = acc[r];
}

// ---------------- attention scores: s[n,c] = x[n].QKs[c] + qb[c] ----------------
__global__ void k_scores(const float* __restrict__ x, const float* __restrict__ qks,
                         const float* __restrict__ qbv, const int* __restrict__ batch,
                         const int* __restrict__ starts, float* __restrict__ sdense) {
  __shared__ float sQK[C_DIM * D_DIM];   // 32 KB
  int t = threadIdx.x;
  for (int i = t; i < C_DIM * D_DIM; i += 128) sQK[i] = qks[i];
  __syncthreads();

  int wid = t >> 5, L = t & 31;
  int n0 = (blockIdx.x * 4 + wid) * 16;
  int m = L & 15, half = L >> 4, kb = half * 2;
  v8f acc0 = {}, acc1 = {};
  const float* xrow = x + (size_t)(n0 + m) * D_DIM;
  for (int kk = 0; kk < D_DIM; kk += 4) {
    v2f a, b0, b1;
    a.x = xrow[kk + kb];
    a.y = xrow[kk + kb + 1];
    b0.x = sQK[m * D_DIM + kk + kb];
    b0.y = sQK[m * D_DIM + kk + kb + 1];
    b1.x = sQK[(16 + m) * D_DIM + kk + kb];
    b1.y = sQK[(16 + m) * D_DIM + kk + kb + 1];
    acc0 = wmma4(a, b0, acc0);
    acc1 = wmma4(a, b1, acc1);
  }
  float q0 = qbv[m], q1 = qbv[16 + m];
  for (int r = 0; r < 8; ++r) {
    int node = n0 + r + half * 8;
    int b = batch[node];
    int l = node - starts[b];              // pos within graph (batch is sorted)
    if ((unsigned)l < (unsigned)L_DIM) {   // drop OOB like the reference scatter
      sdense[((size_t)b * C_DIM + m) * L_DIM + l] = acc0[r] + q0;
      sdense[((size_t)b * C_DIM + 16 + m) * L_DIM + l] = acc1[r] + q1;
    }
  }
}

// ---------------- masked softmax over each (b,c) row ----------------
__global__ void k_softmax(const float* __restrict__ sdense, const int* __restrict__ counts,
                          float* __restrict__ Aout) {
  __shared__ float red[256];
  int row = blockIdx.x;                    // b*32 + c
  int b = row >> 5;
  int t = threadIdx.x;
  int cnt = counts[b]; if (cnt > L_DIM) cnt = L_DIM;
  const float* srow = sdense + (size_t)row * L_DIM;
  float* arow = Aout + (size_t)row * L_DIM;
  if (cnt == 0) {                          // all -1e9 -> uniform softmax
    const float u = 1.0f / (float)L_DIM;
    for (int l = t; l < L_DIM; l += 256) arow[l] = u;
    return;
  }
  float mx = -3.0e38f;
  for (int l = t; l < cnt; l += 256) mx = fmaxf(mx, srow[l]);
  red[t] = mx; __syncthreads();
  for (int off = 128; off > 0; off >>= 1) {
    if (t < off) red[t] = fmaxf(red[t], red[t + off]);
    __syncthreads();
  }
  mx = red[0]; __syncthreads();
  float sum = 0.f;
  for (int l = t; l < cnt; l += 256) sum += __expf(srow[l] - mx);
  red[t] = sum; __syncthreads();
  for (int off = 128; off > 0; off >>= 1) {
    if (t < off) red[t] += red[t + off];
    __syncthreads();
  }
  float inv = 1.0f / red[0];
  for (int l = t; l < L_DIM; l += 256)
    arow[l] = (l < cnt) ? __expf(srow[l] - mx) * inv : 0.0f;  // masked exp underflows to 0
}

__global__ void k_mask(const int* __restrict__ counts, float* __restrict__ maskOut) {
  int i = blockIdx.x * 256 + threadIdx.x;
  int b = i >> 10, l = i & (L_DIM - 1);
  int cnt = counts[b]; if (cnt > L_DIM) cnt = L_DIM;
  maskOut[i] = (l < cnt) ? 1.0f : 0.0f;
}

// ---------------- P[b,c,:] = sum_l A[b,c,l] * x[start+l,:] ----------------
// One 1024-thread block (32 waves) per graph. Per 32-row stage, wave 0 issues
// two TDM tensor_load_to_lds descriptors (x tile 32x256, A tile 32x32),
// double-buffered against TENSORcnt. Rows/cols past the ragged segment end
// are zero-filled by TDM OOB handling, so no staging guards are needed.
__global__ void __launch_bounds__(1024, 1)
k_pool(const float* __restrict__ x, const float* __restrict__ Aout,
       const int* __restrict__ counts, const int* __restrict__ starts,
       float* __restrict__ P) {
  __shared__ float sX[2][32 * D_DIM];     // 2 x 32 KB
  __shared__ float sA[2][C_DIM * 32];     // 2 x 4 KB
  int b = blockIdx.x;
  int t = threadIdx.x;
  int cnt = counts[b]; if (cnt > L_DIM) cnt = L_DIM;
  int start = starts[b];
  cnt = __builtin_amdgcn_readfirstlane(cnt);
  start = __builtin_amdgcn_readfirstlane(start);

  int wid = t >> 5, L = t & 31;
  int ctile = wid & 1, dtile = wid >> 1;
  int m = L & 15, half = L >> 4, kb = half * 2;

  unsigned long long xbase = (unsigned long long)(size_t)x
                           + (unsigned long long)(unsigned)start * (D_DIM * 4ull);
  unsigned long long abase = (unsigned long long)(size_t)Aout
                           + (unsigned long long)(unsigned)b * (C_DIM * L_DIM * 4ull);
  int nstages = (cnt + 31) >> 5;

  if (wid == 0 && nstages > 0) {
    // stage 0: x rows [0,32) -> sX[0]; A cols [0,32) of all 32 clusters -> sA[0]
    tdm_load_2d((unsigned int)(size_t)&sX[0][0], xbase,
                D_DIM, (unsigned)cnt, D_DIM, 32, D_DIM);
    tdm_load_2d((unsigned int)(size_t)&sA[0][0], abase,
                (unsigned)cnt, C_DIM, 32, C_DIM, L_DIM);
  }

  v8f acc = {};
  for (int s = 0; s < nstages; ++s) {
    int buf = s & 1;
    if (wid == 0) {
      if (s + 1 < nstages) {
        int kk0 = (s + 1) * 32;
        tdm_load_2d((unsigned int)(size_t)&sX[buf ^ 1][0],
                    xbase + (unsigned long long)kk0 * (D_DIM * 4ull),
                    D_DIM, (unsigned)(cnt - kk0), D_DIM, 32, D_DIM);
        tdm_load_2d((unsigned int)(size_t)&sA[buf ^ 1][0],
                    abase + (unsigned long long)kk0 * 4ull,
                    (unsigned)(cnt - kk0), C_DIM, 32, C_DIM, L_DIM);
        __builtin_amdgcn_s_wait_tensorcnt(2);   // stage s done; s+1 in flight
      } else {
        __builtin_amdgcn_s_wait_tensorcnt(0);   // last stage: drain
      }
    }
    __syncthreads();                            // stage s tiles visible to all waves
    const float* xs = &sX[buf][0];
    const float* as = &sA[buf][0];
    for (int kk = 0; kk < 32; kk += 4) {
      v2f a, bb;
      a.x = as[(ctile * 16 + m) * 32 + kk + kb];
      a.y = as[(ctile * 16 + m) * 32 + kk + kb + 1];
      bb.x = xs[(kk + kb) * D_DIM + dtile * 16 + m];
      bb.y = xs[(kk + kb + 1) * D_DIM + dtile * 16 + m];
      acc = wmma4(a, bb, acc);
    }
    __syncthreads();                            // buf free before TDM refills it
  }
  for (int r = 0; r < 8; ++r) {
    int c = ctile * 16 + r + half * 8;
    P[((size_t)b * C_DIM + c) * D_DIM + dtile * 16 + m] = acc[r];
  }
}

// ---------------- out[row,:] = relu(W2 @ P[row] + c2[c]) ----------------
__global__ void k_final(const float* __restrict__ P, const float* __restrict__ W2,
                        const float* __restrict__ c2, float* __restrict__ out) {
  __shared__ float sW[16 * D_DIM];        // 16 KB
  int t = threadIdx.x;
  int d0 = blockIdx.x * 16;
  for (int i = t; i < 16 * D_DIM; i += 128) {
    int dr = i >> 8, k = i & 255;
    sW[i] = W2[(size_t)(d0 + dr) * D_DIM + k];
  }
  __syncthreads();
  int wid = t >> 5, L = t & 31;
  int r0 = (blockIdx.y * 4 + wid) * 16;
  int m = L & 15, half = L >> 4, kb = half * 2;
  v8f acc = {};
  const float* prow = P + (size_t)(r0 + m) * D_DIM;
  for (int kk = 0; kk < D_DIM; kk += 4) {
    v2f a, bb;
    a.x = prow[kk + kb];
    a.y = prow[kk + kb + 1];
    bb.x = sW[m * D_DIM + kk + kb];        // B[k][n] = W2[d0+n][k]
    bb.y = sW[m * D_DIM + kk + kb + 1];
    acc = wmma4(a, bb, acc);
  }
  for (int r = 0; r < 8; ++r) {
    int row = r0 + r + half * 8;
    int c = row & (C_DIM - 1);
    float v = acc[r] + c2[c * D_DIM + d0 + m];
    out[(size_t)row * D_DIM + d0 + m] = fmaxf(v, 0.0f);
  }
}

// ---------------- host launch ----------------
extern "C" void kernel_launch(void* const* d_in, const int* in_sizes, int n_in,
                              void* d_out, int out_size, void* d_ws, size_t ws_size,
                              hipStream_t stream) {
  const float* x  = (const float*)d_in[0];
  const int* batch = (const int*)d_in[1];
  const float* Q  = (const float*)d_in[2];
  const float* Wq = (const float*)d_in[3];
  const float* bq = (const float*)d_in[4];
  const float* Wk = (const float*)d_in[5];
  const float* bk = (const float*)d_in[6];
  const float* Wv = (const float*)d_in[7];
  const float* bv = (const float*)d_in[8];
  const float* Wo = (const float*)d_in[9];
  const float* bo = (const float*)d_in[10];

  const int N = in_sizes[0] / D_DIM;      // 131072

  // d_out layout: out [B,C,D] | A [B,C,L] | mask [B,L]
  float* out_p  = (float*)d_out;
  float* A_p    = out_p + (size_t)B_DIM * C_DIM * D_DIM;
  float* mask_p = A_p + (size_t)B_DIM * C_DIM * L_DIM;

  // workspace layout (floats)
  float* ws = (float*)d_ws;
  float* qp     = ws;                     // 8192
  float* qks    = qp + 8192;              // 8192
  float* qbv    = qks + 8192;             // 32 (padded to 256)
  float* c2     = qbv + 256;              // 8192
  float* W2     = c2 + 8192;              // 65536
  int*   counts = (int*)(W2 + 65536);     // 256
  int*   starts = counts + 256;           // 256
  float* sdense = (float*)(starts + 256); // B*C*L = 8388608
  float* P      = sdense;                 // reuse: sdense dead after softmax

  // segment bookkeeping
  k_zero_counts<<<1, 256, 0, stream>>>(counts);
  k_hist<<<(N + 255) / 256, 256, 0, stream>>>(batch, counts, N);
  k_scan<<<1, 256, 0, stream>>>(counts, starts);

  // fused-weight precomputes
  k_qp<<<C_DIM, D_DIM, 0, stream>>>(Q, Wq, bq, qp);
  k_qk<<<C_DIM, D_DIM, 0, stream>>>(qp, Wk, bk, qks, qbv);
  k_c2<<<C_DIM, D_DIM, 0, stream>>>(qp, Wo, bv, bo, c2);
  k_w2<<<dim3(16, 16), 32, 0, stream>>>(Wo, Wv, W2);

  // scores -> softmax(A) -> mask
  k_scores<<<N / 64, 128, 0, stream>>>(x, qks, qbv, batch, starts, sdense);
  k_softmax<<<B_DIM * C_DIM, 256, 0, stream>>>(sdense, counts, A_p);
  k_mask<<<(B_DIM * L_DIM) / 256, 256, 0, stream>>>(counts, mask_p);

  // weighted pooling (TDM-staged) then final projection + relu
  k_pool<<<B_DIM, 1024, 0, stream>>>(x, A_p, counts, starts, P);
  k_final<<<dim3(16, 128), 128, 0, stream>>>(P, W2, c2, out_p);
}